// WaveCell_67147518706351
// MI455X (gfx1250) — compile-verified
//
#include <hip/hip_runtime.h>
#include <stdint.h>

// ---------------- problem geometry ----------------
#define NXg     192
#define NYg     192
#define TSTEPS  256
#define BATCH   4
#define KWG     12                 // row-strip workgroups per batch
#define ROWS    16                 // rows per strip (12*16 = 192)
#define SEGS    48                 // 4-wide column segments per row (48*4 = 192)
#define THREADS (ROWS * SEGS)      // 768 threads = 24 wave32

typedef float    f4  __attribute__((ext_vector_type(4)));
typedef uint32_t v4u __attribute__((ext_vector_type(4)));
typedef int32_t  v8i __attribute__((ext_vector_type(8)));
typedef int32_t  v4i __attribute__((ext_vector_type(4)));

// ---------------- Tensor Data Mover: 1-D f32 tile, LDS -> global ----------------
// D# layout per CDNA5 ISA (2-D descriptor, groups 0+1;
// groups 2/3 unused since tile_dim2 == 0).
__device__ __forceinline__ void tdm_store_tile_f32(uint32_t lds_byte_off,
                                                   uint64_t gaddr,
                                                   uint32_t nwords) {
  v4u g0;
  g0[0] = 0x1u;                                            // count=1, user mode
  g0[1] = lds_byte_off;                                    // lds_addr [63:32]
  g0[2] = (uint32_t)gaddr;                                 // global_addr [95:64]
  g0[3] = ((uint32_t)(gaddr >> 32) & 0x01FFFFFFu)          // global_addr [120:96]
        | (2u << 30);                                      // type=2 ("image")
  v8i g1;
  g1[0] = (int)(2u << 16);                                 // data_size = 4 bytes
  g1[1] = (int)((nwords & 0xFFFFu) << 16);                 // tensor_dim0 lo16 @ [63:48]
  g1[2] = (int)(((nwords >> 16) & 0xFFFFu) | (1u << 16));  // dim0 hi16, tensor_dim1=1
  g1[3] = (int)((nwords & 0xFFFFu) << 16);                 // tile_dim0 @ [127:112]
  g1[4] = 1;                                               // tile_dim1 = 1
  g1[5] = (int)nwords;                                     // tensor_dim0_stride lo32
  g1[6] = 0;
  g1[7] = 0;
  v4i z4 = (v4i)0;
#if defined(__clang_major__) && (__clang_major__ >= 23)
  v8i z8 = (v8i)0;
  __builtin_amdgcn_tensor_store_from_lds(g0, g1, z4, z4, z8, 0);
#else
  __builtin_amdgcn_tensor_store_from_lds(g0, g1, z4, z4, 0);
#endif
}

// ---------------- per-call workspace re-init (replay-safe) ----------------
__global__ void ws_zero_kernel(float* ws, int n) {
  int idx = blockIdx.x * blockDim.x + threadIdx.x;
  int stride = gridDim.x * blockDim.x;
  for (int l = idx; l < n; l += stride) ws[l] = 0.0f;
}

// ---------------- persistent FDTD kernel ----------------
__global__ __launch_bounds__(THREADS, 1)
void wave_persistent_kernel(const float* __restrict__ xin,
                            const float* __restrict__ cmap,
                            const float* __restrict__ bmap,
                            float* __restrict__ out,
                            float* __restrict__ ws) {
  __shared__ __align__(16) float lds[2][ROWS * NYg];   // ping-pong y fields, 24 KB
  __shared__ __align__(16) float xloc[TSTEPS];         // staged source series, 1 KB

  const int tid = threadIdx.x;
  const int b   = blockIdx.x / KWG;
  const int k   = blockIdx.x % KWG;
  const int r0  = k * ROWS;
  const int i   = tid / SEGS;        // 0..15 local row
  const int seg = tid % SEGS;        // 0..47
  const int j0  = seg * 4;
  const int gi  = r0 + i;

  // ws layout (floats): TOP[2][B][KWG][NY], BOT[2][B][KWG][NY], then counters
  float* TOPh = ws;
  float* BOTh = ws + 2 * BATCH * KWG * NYg;
  unsigned* cnt = (unsigned*)(ws + 4 * BATCH * KWG * NYg) + b * 32;

  const float inv_dt2 = 1.0e6f;                                  // 1/DT^2
  const float Hc      = 1.0e-3f * 1.41421356237309515f * 1.01f;  // h
  const float inv_h2  = 1.0f / (Hc * Hc);

  // per-cell coefficients, register-resident
  const f4 cv = *(const f4*)(cmap + gi * NYg + j0);
  const f4 bv = *(const f4*)(bmap + gi * NYg + j0);
  float ainv[4], cf2[4], cc2[4];
#pragma unroll
  for (int c = 0; c < 4; ++c) {
    ainv[c] = 1.0f / (inv_dt2 + 500.0f * bv[c]);   // 0.5*b/DT = 500*b
    cf2[c]  = inv_dt2 - 500.0f * bv[c];
    cc2[c]  = cv[c] * cv[c];
  }

  float ycur[4]  = {0.f, 0.f, 0.f, 0.f};   // y_{t-1} (centers)
  float yprev[4] = {0.f, 0.f, 0.f, 0.f};   // y_{t-2}

  for (int l = tid; l < 2 * ROWS * NYg; l += THREADS) (&lds[0][0])[l] = 0.0f;
  if (tid < TSTEPS) xloc[tid] = xin[b * TSTEPS + tid];   // stage source series
  __syncthreads();

  // point source (96,32): batch-row 96 -> k=6, i=0; col 32 -> seg=8, lane-slot 0
  const bool is_src = (k == 6) && (i == 0) && (seg == 8);

  const uint64_t out_base = (uint64_t)(uintptr_t)out;

#pragma unroll 2              // makes pr/pw compile-time: LDS bases fold to immediates
  for (int t = 0; t < TSTEPS; ++t) {
    const int pr = t & 1;        // y1 buffer (read)
    const int pw = pr ^ 1;       // y_t buffer (write)

    // vertical neighbors (LDS interior, global halo at strip edges)
    f4 upv, dnv;
    if (i > 0)      upv = *(const f4*)&lds[pr][(i - 1) * NYg + j0];
    else if (k > 0) upv = *(const f4*)&BOTh[((pr * BATCH + b) * KWG + (k - 1)) * NYg + j0];
    else            upv = (f4)0.0f;
    if (i < ROWS - 1)     dnv = *(const f4*)&lds[pr][(i + 1) * NYg + j0];
    else if (k < KWG - 1) dnv = *(const f4*)&TOPh[((pr * BATCH + b) * KWG + (k + 1)) * NYg + j0];
    else                  dnv = (f4)0.0f;
    const float lf = (seg > 0)        ? lds[pr][i * NYg + j0 - 1] : 0.0f;
    const float rt = (seg < SEGS - 1) ? lds[pr][i * NYg + j0 + 4] : 0.0f;

    float src = 0.0f;
    if (is_src) src = xloc[t];   // LDS-staged: keeps source off the L2 critical path

    float yn[4];
#pragma unroll
    for (int c = 0; c < 4; ++c) {
      const float lv  = (c > 0) ? ycur[c - 1] : lf;
      const float rv  = (c < 3) ? ycur[c + 1] : rt;
      const float lap = inv_h2 * (upv[c] + dnv[c] + lv + rv - 4.0f * ycur[c]);
      yn[c] = ainv[c] * (2.0f * inv_dt2 * ycur[c] - cf2[c] * yprev[c] + cc2[c] * lap);
    }
    if (is_src) yn[0] += src;

    f4 ynv; ynv[0] = yn[0]; ynv[1] = yn[1]; ynv[2] = yn[2]; ynv[3] = yn[3];
    *(f4*)&lds[pw][i * NYg + j0] = ynv;                                        // ds_store_b128
    if (i == 0)        *(f4*)&TOPh[((pw * BATCH + b) * KWG + k) * NYg + j0] = ynv;
    if (i == ROWS - 1) *(f4*)&BOTh[((pw * BATCH + b) * KWG + k) * NYg + j0] = ynv;

#pragma unroll
    for (int c = 0; c < 4; ++c) { yprev[c] = ycur[c]; ycur[c] = yn[c]; }

    // make halo stores agent-visible, finish all LDS writes of this step
    __builtin_amdgcn_fence(__ATOMIC_RELEASE, "agent");
    __syncthreads();

    if (tid == 0) {
      // async TDM store of this step's strip; overlaps next step's compute.
      const uint64_t gaddr = out_base +
          4ull * ((uint64_t)(b * TSTEPS + t) * (NXg * NYg) + (uint64_t)r0 * NYg);
      tdm_store_tile_f32((uint32_t)(uintptr_t)&lds[pw][0], gaddr, ROWS * NYg);

      // per-batch split barrier across the 12 strip WGs
      __hip_atomic_fetch_add(cnt, 1u, __ATOMIC_RELEASE, __HIP_MEMORY_SCOPE_AGENT);
      const unsigned target = (unsigned)(KWG * (t + 1));
      while (__hip_atomic_load(cnt, __ATOMIC_RELAXED, __HIP_MEMORY_SCOPE_AGENT) < target)
        __builtin_amdgcn_s_sleep(2);

      // buffer written at step t becomes the write target at t+2; leave at most
      // the just-issued TDM outstanding before anyone overwrites LDS.
      __builtin_amdgcn_s_wait_tensorcnt(1);
    }
    __syncthreads();
    __builtin_amdgcn_fence(__ATOMIC_ACQUIRE, "agent");
  }
}

// ---------------- harness entry ----------------
extern "C" void kernel_launch(void* const* d_in, const int* in_sizes, int n_in,
                              void* d_out, int out_size, void* d_ws, size_t ws_size,
                              hipStream_t stream) {
  (void)in_sizes; (void)n_in; (void)out_size; (void)ws_size;
  const float* x  = (const float*)d_in[0];   // [B, T, 1]
  const float* c  = (const float*)d_in[1];   // [192, 192]
  const float* bb = (const float*)d_in[2];   // [192, 192]
  float* out = (float*)d_out;                // [B, T, 192, 192]
  float* ws  = (float*)d_ws;

  // halos (2 parities x TOP/BOT) + barrier counters; must be zeroed every call
  const int zero_n = 4 * BATCH * KWG * NYg + 256;
  ws_zero_kernel<<<dim3(48), dim3(256), 0, stream>>>(ws, zero_n);

  wave_persistent_kernel<<<dim3(BATCH * KWG), dim3(THREADS), 0, stream>>>(
      x, c, bb, out, ws);
}